// GNNPeptideHLAModel_66477503807678
// MI455X (gfx1250) — compile-verified
//
#include <hip/hip_runtime.h>
#include <hip/hip_bf16.h>

#define NN 100000
#define NG 2000
#define HID 64

typedef float v2f __attribute__((ext_vector_type(2)));
typedef float v8f __attribute__((ext_vector_type(8)));

__device__ __forceinline__ void atomAddF(float* p, float v) {
  unsafeAtomicAdd(p, v);   // hardware global_atomic_add_f32 on gfx1250
}

// ---------------- degree / norm ----------------
__global__ void k_fill(float* p, float v, int n) {
  int i = blockIdx.x * blockDim.x + threadIdx.x;
  if (i < n) p[i] = v;
}

__global__ void k_edge_deg(const int* __restrict__ dst, float* __restrict__ deg, int E) {
  int e = blockIdx.x * blockDim.x + threadIdx.x;
  if (e < E) atomAddF(&deg[dst[e]], 1.0f);
}

__global__ void k_dinv(float* __restrict__ dinv, int n) {
  int i = blockIdx.x * blockDim.x + threadIdx.x;
  if (i < n) {
    float d = dinv[i];                 // deg >= 1 always (self loops)
    dinv[i] = rsqrtf(fmaxf(d, 1.0f));
  }
}

// ---------------- WMMA f32 GEMM: out[N x 64] = A[N x K] @ W[K x 64] ----------------
template <int K>
__global__ void k_gemm_wmma(const float* __restrict__ A, const float* __restrict__ W,
                            float* __restrict__ out, int nrows) {
  const int wave = (int)((blockIdx.x * blockDim.x + threadIdx.x) >> 5);
  const int lane = threadIdx.x & 31;
  const int row0 = wave << 4;
  if (row0 >= nrows) return;
  const int m  = lane & 15;
  const int kh = (lane >> 4) << 1;     // 0 or 2
  const float* arow = A + (size_t)(row0 + m) * K;
  v8f acc0 = {}, acc1 = {}, acc2 = {}, acc3 = {};
#pragma unroll
  for (int kb = 0; kb < K; kb += 4) {
    v2f a = *(const v2f*)(arow + kb + kh);          // A[m][kb+kh .. +1]
    const float* wr = W + (size_t)(kb + kh) * HID + m;
    v2f b0, b1, b2, b3;
    b0.x = wr[0];   b0.y = wr[HID];
    b1.x = wr[16];  b1.y = wr[HID + 16];
    b2.x = wr[32];  b2.y = wr[HID + 32];
    b3.x = wr[48];  b3.y = wr[HID + 48];
    acc0 = __builtin_amdgcn_wmma_f32_16x16x4_f32(false, a, false, b0, (short)0, acc0, false, false);
    acc1 = __builtin_amdgcn_wmma_f32_16x16x4_f32(false, a, false, b1, (short)0, acc1, false, false);
    acc2 = __builtin_amdgcn_wmma_f32_16x16x4_f32(false, a, false, b2, (short)0, acc2, false, false);
    acc3 = __builtin_amdgcn_wmma_f32_16x16x4_f32(false, a, false, b3, (short)0, acc3, false, false);
  }
  const int rb = row0 + ((lane >> 4) << 3);
#pragma unroll
  for (int r = 0; r < 8; ++r) {
    float* o = out + (size_t)(rb + r) * HID + m;
    o[0]  = acc0[r];
    o[16] = acc1[r];
    o[32] = acc2[r];
    o[48] = acc3[r];
  }
}

// ---------------- self-loop init: out = dinv[i]^2 * xw ----------------
__global__ void k_selfinit(const float* __restrict__ xw, const float* __restrict__ dinv,
                           float* __restrict__ out, int total) {
  int i = blockIdx.x * blockDim.x + threadIdx.x;
  if (i >= total) return;
  int n = i >> 6;
  float d = dinv[n];
  out[i] = d * d * xw[i];
}

// ---------------- edge scatter: out[dst] += dinv[src]*dinv[dst]*xw[src] ----------------
__global__ void k_scatter(const float* __restrict__ xw, const int* __restrict__ src,
                          const int* __restrict__ dst, const float* __restrict__ dinv,
                          float* __restrict__ out, int total /* = E*16 */) {
  int i = blockIdx.x * blockDim.x + threadIdx.x;
  if (i >= total) return;
  int e = i >> 4;
  int c = (i & 15) << 2;
  int s = src[e], d = dst[e];
  float nrm = dinv[s] * dinv[d];
  const float4 v = *(const float4*)(xw + (size_t)s * HID + c);
  float* o = out + (size_t)d * HID + c;
  atomAddF(o + 0, nrm * v.x);
  atomAddF(o + 1, nrm * v.y);
  atomAddF(o + 2, nrm * v.z);
  atomAddF(o + 3, nrm * v.w);
}

// ---------------- bias + relu in place ----------------
__global__ void k_bias_relu(float* __restrict__ h, const float* __restrict__ b, int total) {
  int i = blockIdx.x * blockDim.x + threadIdx.x;
  if (i >= total) return;
  h[i] = fmaxf(h[i] + b[i & (HID - 1)], 0.0f);
}

// ---------------- pooling ----------------
__global__ void k_pool_sum(const float* __restrict__ h, const int* __restrict__ batch,
                           float* __restrict__ gsum, int total /* = N*16 */) {
  int i = blockIdx.x * blockDim.x + threadIdx.x;
  if (i >= total) return;
  int n = i >> 4;
  int c = (i & 15) << 2;
  int g = batch[n];
  const float4 v = *(const float4*)(h + (size_t)n * HID + c);
  float* o = gsum + (size_t)g * HID + c;
  atomAddF(o + 0, v.x);
  atomAddF(o + 1, v.y);
  atomAddF(o + 2, v.z);
  atomAddF(o + 3, v.w);
}

__global__ void k_pool_cnt(const int* __restrict__ batch, float* __restrict__ gcnt, int n) {
  int i = blockIdx.x * blockDim.x + threadIdx.x;
  if (i < n) atomAddF(&gcnt[batch[i]], 1.0f);
}

// ---------------- head MLP: relu(gm@lw1+lb1) @ lw2 + lb2 ----------------
__global__ void k_head(const float* __restrict__ gsum, const float* __restrict__ gcnt,
                       const float* __restrict__ lw1, const float* __restrict__ lb1,
                       const float* __restrict__ lw2, const float* __restrict__ lb2,
                       float* __restrict__ out) {
  int g = blockIdx.x;
  int j = threadIdx.x;                 // 0..63
  __shared__ float gm[HID];
  __shared__ float t[HID];
  float inv = 1.0f / fmaxf(gcnt[g], 1.0f);
  gm[j] = gsum[(size_t)g * HID + j] * inv;
  __syncthreads();
  float acc = lb1[j];
#pragma unroll
  for (int f = 0; f < HID; ++f) acc += gm[f] * lw1[f * HID + j];
  acc = fmaxf(acc, 0.0f);
  t[j] = acc * lw2[j];
  __syncthreads();
  for (int s = 32; s > 0; s >>= 1) {
    if (j < s) t[j] += t[j + s];
    __syncthreads();
  }
  if (j == 0) out[g] = t[0] + lb2[0];
}

// ---------------- driver ----------------
static inline int cdiv(int a, int b) { return (a + b - 1) / b; }

extern "C" void kernel_launch(void* const* d_in, const int* in_sizes, int n_in,
                              void* d_out, int out_size, void* d_ws, size_t ws_size,
                              hipStream_t stream) {
  const float* x   = (const float*)d_in[0];
  const float* W1  = (const float*)d_in[1];
  const float* b1  = (const float*)d_in[2];
  const float* W2  = (const float*)d_in[3];
  const float* b2  = (const float*)d_in[4];
  const float* W3  = (const float*)d_in[5];
  const float* b3  = (const float*)d_in[6];
  const float* lw1 = (const float*)d_in[7];
  const float* lb1 = (const float*)d_in[8];
  const float* lw2 = (const float*)d_in[9];
  const float* lb2 = (const float*)d_in[10];
  const int* ei    = (const int*)d_in[11];
  const int* batch = (const int*)d_in[12];

  const int N = in_sizes[12];          // 100000
  const int E = in_sizes[11] / 2;      // 1600000
  const int G = NG;
  const int* src = ei;
  const int* dst = ei + E;

  float* ws   = (float*)d_ws;
  float* dinv = ws;                                   // N floats (doubles as deg)
  size_t offA = ((size_t)N + 255) & ~(size_t)255;
  float* bufA = ws + offA;                            // N*64 (GEMM output xw)
  float* bufB = bufA + (size_t)N * HID;               // N*64 (aggregated h)
  float* gsum = bufB + (size_t)N * HID;               // G*64
  float* gcnt = gsum + (size_t)G * HID;               // G

  const int BT = 256;
  const int totNF = N * HID;           // 6.4M
  const int totE4 = E * 16;            // 25.6M
  const int totN4 = N * 16;            // 1.6M
  const int waves = N / 16;            // 6250
  const int gemmBlocks = cdiv(waves * 32, BT);

  // normalization
  k_fill<<<cdiv(N, BT), BT, 0, stream>>>(dinv, 1.0f, N);
  k_edge_deg<<<cdiv(E, BT), BT, 0, stream>>>(dst, dinv, E);
  k_dinv<<<cdiv(N, BT), BT, 0, stream>>>(dinv, N);

  // layer 1 (K = 20)
  k_gemm_wmma<20><<<gemmBlocks, BT, 0, stream>>>(x, W1, bufA, N);
  k_selfinit<<<cdiv(totNF, BT), BT, 0, stream>>>(bufA, dinv, bufB, totNF);
  k_scatter<<<cdiv(totE4, BT), BT, 0, stream>>>(bufA, src, dst, dinv, bufB, totE4);
  k_bias_relu<<<cdiv(totNF, BT), BT, 0, stream>>>(bufB, b1, totNF);

  // layer 2 (K = 64)
  k_gemm_wmma<64><<<gemmBlocks, BT, 0, stream>>>(bufB, W2, bufA, N);
  k_selfinit<<<cdiv(totNF, BT), BT, 0, stream>>>(bufA, dinv, bufB, totNF);
  k_scatter<<<cdiv(totE4, BT), BT, 0, stream>>>(bufA, src, dst, dinv, bufB, totE4);
  k_bias_relu<<<cdiv(totNF, BT), BT, 0, stream>>>(bufB, b2, totNF);

  // layer 3 (K = 64)
  k_gemm_wmma<64><<<gemmBlocks, BT, 0, stream>>>(bufB, W3, bufA, N);
  k_selfinit<<<cdiv(totNF, BT), BT, 0, stream>>>(bufA, dinv, bufB, totNF);
  k_scatter<<<cdiv(totE4, BT), BT, 0, stream>>>(bufA, src, dst, dinv, bufB, totE4);
  k_bias_relu<<<cdiv(totNF, BT), BT, 0, stream>>>(bufB, b3, totNF);

  // global mean pool + head
  k_fill<<<cdiv(G * HID, BT), BT, 0, stream>>>(gsum, 0.0f, G * HID);
  k_fill<<<cdiv(G, BT), BT, 0, stream>>>(gcnt, 0.0f, G);
  k_pool_sum<<<cdiv(totN4, BT), BT, 0, stream>>>(bufB, batch, gsum, totN4);
  k_pool_cnt<<<cdiv(N, BT), BT, 0, stream>>>(batch, gcnt, N);
  k_head<<<G, HID, 0, stream>>>(gsum, gcnt, lw1, lb1, lw2, lb2, (float*)d_out);
}